// GraphSAGEModel_89292370083874
// MI455X (gfx1250) — compile-verified
//
#include <hip/hip_runtime.h>

// GraphSAGE 2-layer forward for MI455X (gfx1250, wave32).
//   out = lin_l(mean_{j in N(i)} x_j) + lin_r(x_i) + b  (x2, ReLU between)
// Aggregation: fp32 global atomics (memory/atomic bound, ~3.2GB + 820M atomics).
// GEMM: V_WMMA_F32_16X16X4_F32 (exact fp32), LDS-staged A tiles, weights via L2.
// Two independent WMMA accumulator chains per wave to hide XDL accumulation latency.

#define HID 256
#define TILE_M 16
#define LDS_STRIDE 260   // 260 % 64 == 4 -> conflict-free float2 fragment reads

typedef float v2f __attribute__((ext_vector_type(2)));
typedef float v8f __attribute__((ext_vector_type(8)));

__global__ void sage_degree_kernel(const int* __restrict__ dst,
                                   float* __restrict__ deg, int num_edges) {
    int e = blockIdx.x * blockDim.x + threadIdx.x;
    if (e < num_edges) atomicAdd(&deg[dst[e]], 1.0f);
}

__global__ void sage_invdeg_kernel(float* __restrict__ deg, int n) {
    int i = blockIdx.x * blockDim.x + threadIdx.x;
    if (i < n) deg[i] = 1.0f / fmaxf(deg[i], 1.0f);
}

// One wave32 per edge: coalesced float4 gather of the 256-float source row,
// per-dword fp32 atomic scatter-add into agg[dst].
__global__ void sage_aggregate_kernel(const float* __restrict__ x,
                                      const int* __restrict__ src,
                                      const int* __restrict__ dst,
                                      float* __restrict__ agg, int num_edges) {
    const int wave = threadIdx.x >> 5;
    const int lane = threadIdx.x & 31;
    const int waves_per_block = blockDim.x >> 5;
    const int e = blockIdx.x * waves_per_block + wave;
    if (e >= num_edges) return;
    const int s = src[e];
    const int d = dst[e];
    const float4* __restrict__ xr = (const float4*)(x + (size_t)s * HID);
    float* __restrict__ ar = agg + (size_t)d * HID;
#pragma unroll
    for (int half = 0; half < 2; ++half) {
        float4 v = xr[half * 32 + lane];
        const int base = half * 128 + lane * 4;
        atomicAdd(&ar[base + 0], v.x);
        atomicAdd(&ar[base + 1], v.y);
        atomicAdd(&ar[base + 2], v.z);
        atomicAdd(&ar[base + 3], v.w);
    }
}

// Block: 512 threads = 16 waves. Block owns 16 output rows (m0..m0+15),
// wave w owns columns [16w, 16w+16). Each wave accumulates
//   D = A1(16xK) * Wl^T(Kx16) + A2(16xK) * Wr^T(Kx16)
// with 2 * (256/4) = 128 V_WMMA_F32_16X16X4_F32 per tile, split across
// two independent accumulator chains (accL, accR) for ILP.
template <bool RELU>
__global__ __launch_bounds__(512)
void sage_gemm_kernel(const float* __restrict__ agg,     // un-normalized nbr sums
                      const float* __restrict__ invdeg,  // 1/max(deg,1)
                      const float* __restrict__ x,       // self features
                      const float* __restrict__ wl,      // [O,K] row-major
                      const float* __restrict__ wr,      // [O,K] row-major
                      const float* __restrict__ bias,    // [O]
                      float* __restrict__ out) {
    __shared__ float sA1[TILE_M * LDS_STRIDE];
    __shared__ float sA2[TILE_M * LDS_STRIDE];

    const int m0 = blockIdx.x * TILE_M;
    const int tid = threadIdx.x;

    // Stage both 16x256 A tiles into LDS; pre-scale the neighbor sums by invdeg.
    // 16*256 = 4096 floats = 1024 float4 slots; 512 threads -> 2 slots each.
#pragma unroll
    for (int it = 0; it < 2; ++it) {
        const int slot = it * 512 + tid;       // 0..1023
        const int row = slot >> 6;             // 64 float4 per row
        const int c4 = (slot & 63) << 2;       // float column
        const float sc = invdeg[m0 + row];
        const float4 a1 = *(const float4*)(agg + (size_t)(m0 + row) * HID + c4);
        const float4 a2 = *(const float4*)(x + (size_t)(m0 + row) * HID + c4);
        float* p1 = &sA1[row * LDS_STRIDE + c4];
        float* p2 = &sA2[row * LDS_STRIDE + c4];
        p1[0] = a1.x * sc; p1[1] = a1.y * sc; p1[2] = a1.z * sc; p1[3] = a1.w * sc;
        p2[0] = a2.x;      p2[1] = a2.y;      p2[2] = a2.z;      p2[3] = a2.w;
    }
    __syncthreads();

    const int wave = tid >> 5;       // 0..15 -> 16-column slice
    const int lane = tid & 31;
    const int n0 = wave * 16;
    const int r = lane & 15;         // A row (lanes 0-15/16-31) and B column
    const int ko = (lane >> 4) << 1; // lanes 16-31 hold K=k+2,k+3 (ISA layout)

    const float* __restrict__ wlp = wl + (size_t)(n0 + r) * HID;  // B = W^T: K-contiguous
    const float* __restrict__ wrp = wr + (size_t)(n0 + r) * HID;
    const float* a1p = &sA1[r * LDS_STRIDE];
    const float* a2p = &sA2[r * LDS_STRIDE];

    v8f accL = {};   // nbr_mean @ Wl^T chain
    v8f accR = {};   // x @ Wr^T chain (independent -> interleaved issue)
#pragma unroll 4
    for (int k = 0; k < HID; k += 4) {
        const v2f a1 = *(const v2f*)(a1p + k + ko);
        const v2f b1 = *(const v2f*)(wlp + k + ko);
        accL = __builtin_amdgcn_wmma_f32_16x16x4_f32(false, a1, false, b1,
                                                     (short)0, accL, false, false);
        const v2f a2 = *(const v2f*)(a2p + k + ko);
        const v2f b2 = *(const v2f*)(wrp + k + ko);
        accR = __builtin_amdgcn_wmma_f32_16x16x4_f32(false, a2, false, b2,
                                                     (short)0, accR, false, false);
    }

    // C/D layout: VGPR rr -> rows rr (lanes 0-15) and rr+8 (lanes 16-31),
    // lane&15 -> column. Fuse chain-merge + bias (+ ReLU).
    const int col = n0 + r;
    const float bv = bias[col];
    const int mbase = m0 + ((lane >> 4) << 3);
#pragma unroll
    for (int rr = 0; rr < 8; ++rr) {
        float v = accL[rr] + accR[rr] + bv;
        if (RELU) v = fmaxf(v, 0.0f);
        out[(size_t)(mbase + rr) * HID + col] = v;
    }
}

extern "C" void kernel_launch(void* const* d_in, const int* in_sizes, int n_in,
                              void* d_out, int out_size, void* d_ws, size_t ws_size,
                              hipStream_t stream) {
    const float* x   = (const float*)d_in[0];
    const float* w1l = (const float*)d_in[1];
    const float* w1r = (const float*)d_in[2];
    const float* b1  = (const float*)d_in[3];
    const float* w2l = (const float*)d_in[4];
    const float* w2r = (const float*)d_in[5];
    const float* b2  = (const float*)d_in[6];
    const int*   ei  = (const int*)d_in[7];

    const int N = in_sizes[0] / HID;   // 100000 (divisible by 16)
    const int E = in_sizes[7] / 2;     // 1600000
    const int* src = ei;
    const int* dst = ei + E;

    // Workspace: agg[N*H] | h[N*H] | deg[N]  (~205 MB)
    float* agg = (float*)d_ws;
    float* h   = agg + (size_t)N * HID;
    float* deg = h + (size_t)N * HID;

    const int agg_blocks = (E + 7) / 8;     // 8 edge-waves per 256-thread block
    const int gemm_blocks = N / TILE_M;     // 6250

    // Degree / inverse-degree (shared by both layers: same edge set).
    hipMemsetAsync(deg, 0, (size_t)N * sizeof(float), stream);
    sage_degree_kernel<<<(E + 255) / 256, 256, 0, stream>>>(dst, deg, E);
    sage_invdeg_kernel<<<(N + 255) / 256, 256, 0, stream>>>(deg, N);

    // Layer 1: aggregate(x) -> GEMM (+bias, ReLU) -> h
    hipMemsetAsync(agg, 0, (size_t)N * HID * sizeof(float), stream);
    sage_aggregate_kernel<<<agg_blocks, 256, 0, stream>>>(x, src, dst, agg, E);
    sage_gemm_kernel<true><<<gemm_blocks, 512, 0, stream>>>(agg, deg, x, w1l, w1r, b1, h);

    // Layer 2: aggregate(h) -> GEMM (+bias) -> out
    hipMemsetAsync(agg, 0, (size_t)N * HID * sizeof(float), stream);
    sage_aggregate_kernel<<<agg_blocks, 256, 0, stream>>>(h, src, dst, agg, E);
    sage_gemm_kernel<false><<<gemm_blocks, 512, 0, stream>>>(agg, deg, h, w2l, w2r, b2,
                                                             (float*)d_out);
}